// HybridMemory_72430328480031
// MI455X (gfx1250) — compile-verified
//
#include <hip/hip_runtime.h>
#include <hip/hip_bf16.h>
#include <stdint.h>

#define MOMENTUM 0.2f
#define BATCH    8192
#define NF       1024
#define NS       50000
#define MAXCHAIN 64

// ---------------------------------------------------------------------------
// Kernel 1: bulk bank copy (features -> out), 204.8 MB, via CDNA5 async DMA.
// Each thread owns a private 64B LDS staging region (4 x 16B slots).
// 12,500 blocks x 256 threads x 4 slots x 16B == 204,800,000 B exactly.
// ASYNCcnt tracks the DMA; s_wait_asynccnt 0 orders load->store per wave
// (each lane stores exactly the LDS bytes its own loads produced, so no
// cross-wave barrier is needed). Latency of the two waits is covered by the
// ~100K resident waves.
// ---------------------------------------------------------------------------
__global__ void __launch_bounds__(256) hm_copy_async(const float4* __restrict__ src,
                                                     float4* __restrict__ dst) {
  __shared__ __align__(16) unsigned char smem[256 * 64];
  const unsigned tid   = threadIdx.x;
  const unsigned lbase = (unsigned)(uintptr_t)(&smem[tid * 64]);  // LDS byte offset
  const long Tn = (long)gridDim.x * 256;                 // 3,200,000 threads
  const long T  = (long)blockIdx.x * 256 + tid;

#pragma unroll
  for (int s = 0; s < 4; ++s) {
    const float4* g = src + (long)s * Tn + T;            // lanes contiguous per slot
    unsigned l = lbase + (unsigned)(s * 16);
    asm volatile("global_load_async_to_lds_b128 %0, %1, off"
                 :: "v"(l), "v"(g) : "memory");
  }
  asm volatile("s_wait_asynccnt 0x0" ::: "memory");

#pragma unroll
  for (int s = 0; s < 4; ++s) {
    float4* g = dst + (long)s * Tn + T;
    unsigned l = lbase + (unsigned)(s * 16);
    asm volatile("global_store_async_from_lds_b128 %0, %1, off"
                 :: "v"(g), "v"(l) : "memory");
  }
  asm volatile("s_wait_asynccnt 0x0" ::: "memory");
}

// ---------------------------------------------------------------------------
// Kernel 2: chain-parallel momentum update preserving lax.scan semantics.
// Block i owns label y = labels[i] iff i is the first occurrence of y; the
// owner folds every occurrence j of y in increasing batch order:
//   v = m*v + (1-m)*f_out[j];  v /= ||v||_2
// Occurrences are collected cooperatively (256 threads x 32-stride scan of
// the L2-resident 32KB label vector) into a tiny shared list, sorted, and
// used directly — no serial 8192-iteration scans. 256 threads x float4
// covers the 1024-wide row; wave32 shfl_xor + 8-partial LDS reduction
// computes ||v||^2 per occurrence. All branches containing __syncthreads
// or the block-wide exit are uniform (depend only on shared data/blockIdx).
// ---------------------------------------------------------------------------
__global__ void __launch_bounds__(256) hm_update(const float* __restrict__ bank,
                                                 const float* __restrict__ f_out,
                                                 const int*   __restrict__ labels,
                                                 float* __restrict__ out) {
  __shared__ int   s_match[MAXCHAIN];
  __shared__ int   s_cnt;
  __shared__ float s_part[8];
  __shared__ float s_total;

  const int tid = threadIdx.x;
  const int i   = blockIdx.x;
  const int y   = labels[i];            // uniform scalar load

  if (tid == 0) s_cnt = 0;
  __syncthreads();

  // Cooperative occurrence collection: thread t scans j = t, t+256, ...
  for (int j = tid; j < BATCH; j += 256) {
    if (labels[j] == y) {
      int k = atomicAdd(&s_cnt, 1);
      if (k < MAXCHAIN) s_match[k] = j;
    }
  }
  __syncthreads();

  const int cnt = (s_cnt < MAXCHAIN) ? s_cnt : MAXCHAIN;
  if (tid == 0) {
    // Deterministic order: insertion sort (cnt is ~1-5 in practice).
    for (int a = 1; a < cnt; ++a) {
      int vj = s_match[a];
      int b  = a - 1;
      while (b >= 0 && s_match[b] > vj) { s_match[b + 1] = s_match[b]; --b; }
      s_match[b + 1] = vj;
    }
  }
  __syncthreads();

  // Ownership: i must be the minimum matching index (first occurrence).
  if (s_match[0] != i) return;          // uniform exit

  const long rowy = (long)y * NF;
  float4 v = *(const float4*)(bank + rowy + (long)tid * 4);

  const float m  = MOMENTUM;
  const float om = 1.0f - MOMENTUM;

  for (int q = 0; q < cnt; ++q) {
    const int j = s_match[q];
    if (q + 1 < cnt)                    // warm L2/L0 for the next chain link
      __builtin_prefetch(f_out + (long)s_match[q + 1] * NF + (long)tid * 4, 0, 0);

    float4 x = *(const float4*)(f_out + (long)j * NF + (long)tid * 4);
    v.x = m * v.x + om * x.x;
    v.y = m * v.y + om * x.y;
    v.z = m * v.z + om * x.z;
    v.w = m * v.w + om * x.w;

    float ss = v.x * v.x + v.y * v.y + v.z * v.z + v.w * v.w;
#pragma unroll
    for (int off = 16; off > 0; off >>= 1)
      ss += __shfl_xor(ss, off, 32);    // wave32 reduction
    if ((tid & 31) == 0) s_part[tid >> 5] = ss;
    __syncthreads();
    if (tid == 0) {
      float t = 0.0f;
#pragma unroll
      for (int w = 0; w < 8; ++w) t += s_part[w];
      s_total = t;
    }
    __syncthreads();

    const float inv = 1.0f / sqrtf(s_total);
    v.x *= inv; v.y *= inv; v.z *= inv; v.w *= inv;
  }

  *(float4*)(out + rowy + (long)tid * 4) = v;
}

// ---------------------------------------------------------------------------
// Inputs (setup_inputs order): d_in[0]=f_out [8192,1024] f32,
// d_in[1]=features [50000,1024] f32, d_in[2]=p_labels [8192] int.
// d_out: updated bank [50000,1024] f32.
// ---------------------------------------------------------------------------
extern "C" void kernel_launch(void* const* d_in, const int* in_sizes, int n_in,
                              void* d_out, int out_size, void* d_ws, size_t ws_size,
                              hipStream_t stream) {
  const float* f_out    = (const float*)d_in[0];
  const float* features = (const float*)d_in[1];
  const int*   labels   = (const int*)d_in[2];
  float*       out      = (float*)d_out;

  // 204.8 MB copy: 12,500 blocks * 256 thr * 64 B = exact, no tail.
  hm_copy_async<<<12500, 256, 0, stream>>>((const float4*)features, (float4*)out);

  // One block per batch index; owners fold their whole duplicate-label chain.
  hm_update<<<BATCH, 256, 0, stream>>>(features, f_out, labels, out);
}